// DnlsLoss_70196945486281
// MI455X (gfx1250) — compile-verified
//
#include <hip/hip_runtime.h>
#include <math.h>

// Problem constants (reference)
#define TT   5
#define CC   3
#define HH   128
#define WW   128
#define HW   (HH*WW)
#define WS_  9
#define W2   81          // WS*WS
#define KK   10
#define PSD  7           // PS_DISTS
#define NQ   (TT*HH*WW)  // 81920 queries (B=1)

// Launch config
#define BLK           256
#define WAVES_PER_BLK 8
#define NBLK          1280
#define NWAVES        (NBLK*WAVES_PER_BLK)  // 10240
#define ITERS         (NQ/NWAVES)           // 8, exact

typedef float v2f __attribute__((ext_vector_type(2)));
typedef float v8f __attribute__((ext_vector_type(8)));

static __device__ __forceinline__ int iclamp(int v, int lo, int hi) {
  return v < lo ? lo : (v > hi ? hi : v);
}

// --- Kernel 0: compute patch size from curr_epoch (device-resident scalar) ---
__global__ void ps_kernel(const int* __restrict__ ep, int* __restrict__ ps_out) {
  int step = ep[0];
  double a = pow(0.99993, (double)step);          // PS_SCALE ** step
  double v = a * 7.0 + (1.0 - a) * 1.0;           // a*PS0 + (1-a)*PS_FINAL
  int ps = (int)rint(v);                          // round-half-even == Python round()
  if (ps < 1) ps = 1;                             // max(ps, PS_FINAL)
  if ((ps & 1) == 0) ps += 1;                     // force odd
  ps_out[0] = ps;
}

// --- Kernel 1: per-query search + topK + refine; one wave32 per query ---
__global__ void __launch_bounds__(BLK) nls_main(
    const float* __restrict__ deno, const float* __restrict__ noisy,
    const float* __restrict__ fflow, const float* __restrict__ bflow,
    const int* __restrict__ psp, float* __restrict__ partials)
{
  __shared__ float s_qp  [WAVES_PER_BLK][CC*PSD*PSD]; // 7x7x3 deno query patch
  __shared__ float s_dist[WAVES_PER_BLK][2*W2];       // 162 candidate distances
  __shared__ int   s_sel [WAVES_PER_BLK][2*KK];
  __shared__ float s_seld[WAVES_PER_BLK][2*KK];
  __shared__ float s_bsum[WAVES_PER_BLK];

  const int   ps       = psp[0];
  const int   psh      = ps >> 1;
  const float inv_norm = 1.0f / (float)(ps * ps * CC);

  const int lane = threadIdx.x & 31;
  const int wvb  = threadIdx.x >> 5;
  const int wave = blockIdx.x * WAVES_PER_BLK + wvb;

  float waveAcc = 0.0f;

  for (int it = 0; it < ITERS; ++it) {
    const int q  = wave + it * NWAVES;
    const int qt = q / HW;
    const int rr = q - qt * HW;
    const int qy = rr >> 7;     // /128
    const int qx = rr & 127;

    // Load clipped 7x7x3 query patch from deno into LDS (shared by the wave).
    for (int i = lane; i < CC * 49; i += 32) {
      int c  = i / 49;
      int j  = i - c * 49;
      int dy = j / 7 - 3, dx = j % 7 - 3;
      int yy = iclamp(qy + dy, 0, HH - 1);
      int xx = iclamp(qx + dx, 0, WW - 1);
      s_qp[wvb][i] = deno[(qt * CC + c) * HW + yy * WW + xx];
    }

    // Flow-shifted search centers (jnp.round == round-half-even == rintf).
    const int ffy = (int)rintf(fflow[(qt * 2 + 1) * HW + rr]);
    const int ffx = (int)rintf(fflow[(qt * 2 + 0) * HW + rr]);
    const int bfy = (int)rintf(bflow[(qt * 2 + 1) * HW + rr]);
    const int bfx = (int)rintf(bflow[(qt * 2 + 0) * HW + rr]);
    const int cenY[2]  = { qy + ffy, qy + bfy };
    const int cenX[2]  = { qx + ffx, qx + bfx };
    const int ctArr[2] = { iclamp(qt + 1, 0, TT - 1), iclamp(qt - 1, 0, TT - 1) };

    __syncthreads(); // qpatch visible (all waves in block have identical trip counts)

    // Search distances: lane-per-candidate over the 81-wide window, both directions.
    for (int s = 0; s < 2; ++s) {
      const float* __restrict__ dbase = deno + ctArr[s] * CC * HW;
      for (int w = lane; w < W2; w += 32) {
        int cy0 = iclamp(cenY[s] + (w / WS_) - 4, 0, HH - 1);
        int cx0 = iclamp(cenX[s] + (w % WS_) - 4, 0, WW - 1);
        float acc = 0.0f;
        for (int dy = -psh; dy <= psh; ++dy) {
          int yy   = iclamp(cy0 + dy, 0, HH - 1);
          int qrow = (dy + 3) * 7;
          for (int dx = -psh; dx <= psh; ++dx) {
            int xx = iclamp(cx0 + dx, 0, WW - 1);
            int qi = qrow + dx + 3;
            int gb = yy * WW + xx;
#pragma unroll
            for (int c = 0; c < CC; ++c) {
              float d = s_qp[wvb][c * 49 + qi] - dbase[c * HW + gb];
              acc = fmaf(d, d, acc);
            }
          }
        }
        s_dist[wvb][s * W2 + w] = acc;
      }
    }
    __syncthreads();

    // Top-K smallest per direction: iterative wave32 argmin, tie -> lowest index
    // (matches lax.top_k stability on the negated distances).
    for (int s = 0; s < 2; ++s) {
      for (int k = 0; k < KK; ++k) {
        float bv = __builtin_inff();
        int   bi = W2;
        for (int w = lane; w < W2; w += 32) {
          float v = s_dist[wvb][s * W2 + w];
          if (v < bv) { bv = v; bi = w; }   // ascending scan keeps lowest w on ties
        }
        for (int off = 16; off > 0; off >>= 1) {
          float ov = __shfl_xor(bv, off, 32);
          int   oi = __shfl_xor(bi, off, 32);
          if (ov < bv || (ov == bv && oi < bi)) { bv = ov; bi = oi; }
        }
        if (lane == 0) {
          s_sel [wvb][s * KK + k] = bi;
          s_seld[wvb][s * KK + k] = bv;
          s_dist[wvb][s * W2 + bi] = __builtin_inff();
        }
        __syncthreads();
      }
    }

    // Refine: lanes 0..19 handle the 20 selections (deno query vs noisy candidate, 7x7).
    float contrib = 0.0f;
    if (lane < 2 * KK) {
      int   s    = lane / KK;
      int   widx = s_sel[wvb][lane];
      float d0   = s_seld[wvb][lane];
      float mask = (d0 * inv_norm < 0.5f) ? 1.0f : 0.0f;
      int cy0 = iclamp(cenY[s] + widx / WS_ - 4, 0, HH - 1);
      int cx0 = iclamp(cenX[s] + widx % WS_ - 4, 0, WW - 1);
      const float* __restrict__ nbase = noisy + ctArr[s] * CC * HW;
      float dref = 0.0f;
      for (int dy = -3; dy <= 3; ++dy) {
        int yy   = iclamp(cy0 + dy, 0, HH - 1);
        int qrow = (dy + 3) * 7;
        for (int dx = -3; dx <= 3; ++dx) {
          int xx = iclamp(cx0 + dx, 0, WW - 1);
          int qi = qrow + dx + 3;
          int gb = yy * WW + xx;
#pragma unroll
          for (int c = 0; c < CC; ++c) {
            float d = s_qp[wvb][c * 49 + qi] - nbase[c * HW + gb];
            dref = fmaf(d, d, dref);
          }
        }
      }
      contrib = mask * dref;
    }
    for (int off = 16; off > 0; off >>= 1) contrib += __shfl_xor(contrib, off, 32);
    if (lane == 0) waveAcc += contrib;
    __syncthreads(); // protect s_qp/s_sel before next iteration overwrites them
  }

  if (lane == 0) s_bsum[wvb] = waveAcc;
  __syncthreads();
  if (threadIdx.x == 0) {
    float t = 0.0f;
    for (int i = 0; i < WAVES_PER_BLK; ++i) t += s_bsum[i];
    partials[blockIdx.x] = t; // every block writes its slot -> no ws pre-zeroing needed
  }
}

// --- Kernel 2: WMMA-based reduction of 1280 partials -> mean ---
// D = ones(16x4) x B(4x16) + C accumulates exact f32 column sums; summing all of
// D row 0 yields the total of every B element regardless of operand lane layout.
__global__ void reduce_finalize(const float* __restrict__ partials, float* __restrict__ out)
{
  const int lane = threadIdx.x;      // launched with exactly 32 threads: EXEC all-ones
  v2f a; a.x = 1.0f; a.y = 1.0f;     // A = ones (16x4 f32 = 2 VGPRs/lane)
  v8f acc = {};
  for (int chunk = 0; chunk < NBLK / 64; ++chunk) {   // 20 chunks of 64 values
    v2f b;
    b.x = partials[chunk * 64 + lane];
    b.y = partials[chunk * 64 + 32 + lane];
    acc = __builtin_amdgcn_wmma_f32_16x16x4_f32(
        /*neg_a=*/false, a, /*neg_b=*/false, b,
        /*c_mod=*/(short)0, acc, /*reuse_a=*/false, /*reuse_b=*/false);
  }
  float v = (lane < 16) ? acc[0] : 0.0f;   // row M=0 lives in VGPR0 of lanes 0..15
  for (int off = 16; off > 0; off >>= 1) v += __shfl_xor(v, off, 32);
  if (lane == 0) out[0] = v / (float)(NQ * 2 * KK);  // mean over [Q, S*K], B=1
}

extern "C" void kernel_launch(void* const* d_in, const int* in_sizes, int n_in,
                              void* d_out, int out_size, void* d_ws, size_t ws_size,
                              hipStream_t stream) {
  (void)in_sizes; (void)n_in; (void)out_size; (void)ws_size;
  const float* noisy = (const float*)d_in[0];
  const float* clean = (const float*)d_in[1]; (void)clean;   // unused by loss
  const float* deno  = (const float*)d_in[2];                // srch == deno
  const float* fflow = (const float*)d_in[3];
  const float* bflow = (const float*)d_in[4];
  const int*   epoch = (const int*)d_in[5];

  float* partials = (float*)d_ws;                         // NBLK floats
  int*   ps_slot  = (int*)((char*)d_ws + 16384);          // past partials

  ps_kernel<<<1, 1, 0, stream>>>(epoch, ps_slot);
  nls_main<<<NBLK, BLK, 0, stream>>>(deno, noisy, fflow, bflow, ps_slot, partials);
  reduce_finalize<<<1, 32, 0, stream>>>(partials, (float*)d_out);
}